// FCGAT_17291538334193
// MI455X (gfx1250) — compile-verified
//
#include <hip/hip_runtime.h>
#include <hip/hip_bf16.h>
#include <math.h>

#define N_NODES   100000
#define N_EDGES   1600000
#define F_IN      100
#define HEADS     3
#define HEAD_DIM  64
#define HIDDEN    192
#define NUM_GRAPHS 128
#define NUM_CLASSES 2
#define E_TOT     (N_EDGES + N_NODES)   // self-loops appended
#define SLOPE_ATT 0.2f
#define SLOPE_ACT 0.01f

typedef __attribute__((ext_vector_type(2))) float v2f;
typedef __attribute__((ext_vector_type(8))) float v8f;

// ---- workspace layout (float offsets) ----
#define WS_H      0
#define WS_OUT    (WS_H    + N_NODES * HIDDEN)
#define WS_ASRC   (WS_OUT  + N_NODES * HIDDEN)
#define WS_ADST   (WS_ASRC + N_NODES * HEADS)
#define WS_DENOM  (WS_ADST + N_NODES * HEADS)
#define WS_POOL   (WS_DENOM + N_NODES * HEADS)

// ---------------------------------------------------------------------------
// 0) init: zero atomic accumulators, -inf for max-pool
// ---------------------------------------------------------------------------
__global__ void gat_init(float* ws) {
    long i = (long)blockIdx.x * blockDim.x + threadIdx.x;
    long stride = (long)gridDim.x * blockDim.x;
    for (long j = i; j < (long)N_NODES * HIDDEN; j += stride) ws[WS_OUT + j] = 0.0f;
    for (long j = i; j < (long)N_NODES * HEADS;  j += stride) ws[WS_DENOM + j] = 0.0f;
    for (long j = i; j < (long)NUM_GRAPHS * HIDDEN; j += stride) ws[WS_POOL + j] = -INFINITY;
}

// ---------------------------------------------------------------------------
// 1) h = x @ W via V_WMMA_F32_16X16X4_F32. One wave per 16x16 output tile.
//    grid = (6250 row tiles, 12 col tiles), block = 32 (full wave, EXEC all 1s)
// ---------------------------------------------------------------------------
__global__ __launch_bounds__(32) void gat_gemm_wmma(const float* __restrict__ x,
                                                    const float* __restrict__ W,
                                                    float* __restrict__ ws) {
    float* hbuf = ws + WS_H;
    const int lane = threadIdx.x;
    const int m    = lane & 15;          // A: row M; B: col N
    const int kh   = (lane >> 4) << 1;   // K sub-offset: 0 (lanes 0-15) or 2 (16-31)

    const int row0 = blockIdx.x * 16;
    const int col0 = blockIdx.y * 16;

    const float* xrow = x + (long)(row0 + m) * F_IN;   // 16B-aligned rows (400B stride)
    const float* wcol = W + (col0 + m);

    v8f acc = {};
#pragma unroll 5
    for (int k0 = 0; k0 < F_IN; k0 += 4) {
        v2f a = *(const v2f*)(xrow + k0 + kh);         // K = k0+kh, k0+kh+1
        v2f b;
        b.x = wcol[(long)(k0 + kh) * HIDDEN];
        b.y = wcol[(long)(k0 + kh + 1) * HIDDEN];
        acc = __builtin_amdgcn_wmma_f32_16x16x4_f32(
            /*neg_a=*/false, a, /*neg_b=*/false, b,
            /*c_mod=*/(short)0, acc, /*reuse_a=*/false, /*reuse_b=*/false);
    }

    // D layout: lane l, VGPR r -> (M = r + (l>=16 ? 8 : 0), N = l%16)
    const int mbase = row0 + ((lane >> 4) << 3);
    const int ncol  = col0 + m;
#pragma unroll
    for (int r = 0; r < 8; ++r)
        hbuf[(long)(mbase + r) * HIDDEN + ncol] = acc[r];
}

// ---------------------------------------------------------------------------
// 2) per-(node,head) attention logits a_src, a_dst
// ---------------------------------------------------------------------------
__global__ void gat_logits(const float* __restrict__ att_src,
                           const float* __restrict__ att_dst,
                           float* __restrict__ ws) {
    const float* hbuf = ws + WS_H;
    int i = blockIdx.x * blockDim.x + threadIdx.x;   // (n*HEADS + hd)
    if (i >= N_NODES * HEADS) return;
    int n = i / HEADS, hd = i % HEADS;
    const float* hp = hbuf + (long)n * HIDDEN + hd * HEAD_DIM;
    const float* as = att_src + hd * HEAD_DIM;
    const float* ad = att_dst + hd * HEAD_DIM;
    float s1 = 0.0f, s2 = 0.0f;
#pragma unroll 8
    for (int c = 0; c < HEAD_DIM; ++c) {
        float v = hp[c];
        s1 += v * as[c];
        s2 += v * ad[c];
    }
    ws[WS_ASRC + i] = s1;
    ws[WS_ADST + i] = s2;
}

// ---------------------------------------------------------------------------
// 3) softmax denominators: denom[dst,h] += exp(leakyrelu(a_src[src]+a_dst[dst]))
//    (segment-max shift skipped: softmax is shift-invariant and |e| is small)
// ---------------------------------------------------------------------------
__global__ void gat_denom(const int* __restrict__ ei, float* __restrict__ ws) {
    int e = blockIdx.x * blockDim.x + threadIdx.x;
    if (e >= E_TOT) return;
    int src, dst;
    if (e < N_EDGES) { src = ei[e]; dst = ei[N_EDGES + e]; }
    else             { src = dst = e - N_EDGES; }
#pragma unroll
    for (int hd = 0; hd < HEADS; ++hd) {
        float v = ws[WS_ASRC + src * HEADS + hd] + ws[WS_ADST + dst * HEADS + hd];
        v = (v > 0.0f) ? v : SLOPE_ATT * v;
        atomicAdd(&ws[WS_DENOM + dst * HEADS + hd], __expf(v));
    }
}

// ---------------------------------------------------------------------------
// 4) message scatter: out[dst] += alpha * h[src]. One wave per edge;
//    lanes 0-2 compute head alphas, broadcast, 6 coalesced atomic sweeps.
// ---------------------------------------------------------------------------
__global__ void gat_scatter(const int* __restrict__ ei, float* __restrict__ ws) {
    int gid  = blockIdx.x * blockDim.x + threadIdx.x;
    int e    = gid >> 5;
    int lane = gid & 31;
    if (e >= E_TOT) return;
    int src, dst;
    if (e < N_EDGES) { src = ei[e]; dst = ei[N_EDGES + e]; }
    else             { src = dst = e - N_EDGES; }

    float alpha = 0.0f;
    if (lane < HEADS) {
        float v = ws[WS_ASRC + src * HEADS + lane] + ws[WS_ADST + dst * HEADS + lane];
        v = (v > 0.0f) ? v : SLOPE_ATT * v;
        alpha = __expf(v) / ws[WS_DENOM + dst * HEADS + lane];
    }
    float al0 = __shfl(alpha, 0, 32);
    float al1 = __shfl(alpha, 1, 32);
    float al2 = __shfl(alpha, 2, 32);

    const float* hs = ws + WS_H   + (long)src * HIDDEN;
    float*       op = ws + WS_OUT + (long)dst * HIDDEN;
#pragma unroll
    for (int j = 0; j < HIDDEN / 32; ++j) {
        int c = lane + j * 32;
        float a_c = (c < HEAD_DIM) ? al0 : ((c < 2 * HEAD_DIM) ? al1 : al2);
        atomicAdd(&op[c], hs[c] * a_c);
    }
}

// ---------------------------------------------------------------------------
// 5) bias + leaky_relu + global max pool (float atomicMax bit-trick)
// ---------------------------------------------------------------------------
__device__ __forceinline__ void atomicMaxF32(float* addr, float v) {
    if (v >= 0.0f) atomicMax((int*)addr, __float_as_int(v));
    else           atomicMin((unsigned int*)addr, __float_as_uint(v));
}

__global__ void gat_pool(const float* __restrict__ bias,
                         const int* __restrict__ batch,
                         float* __restrict__ ws) {
    long i = (long)blockIdx.x * blockDim.x + threadIdx.x;   // n*HIDDEN + c
    if (i >= (long)N_NODES * HIDDEN) return;
    int n = (int)(i / HIDDEN), c = (int)(i % HIDDEN);
    float v = ws[WS_OUT + i] + bias[c];
    v = (v > 0.0f) ? v : SLOPE_ACT * v;
    atomicMaxF32(&ws[WS_POOL + (long)batch[n] * HIDDEN + c], v);
}

// ---------------------------------------------------------------------------
// 6) classifier: pooled[128,192] @ cls_W[192,2] + cls_b
// ---------------------------------------------------------------------------
__global__ void gat_classify(const float* __restrict__ cls_W,
                             const float* __restrict__ cls_b,
                             const float* __restrict__ ws,
                             float* __restrict__ out) {
    int t = blockIdx.x * blockDim.x + threadIdx.x;
    if (t >= NUM_GRAPHS * NUM_CLASSES) return;
    int g = t / NUM_CLASSES, k = t % NUM_CLASSES;
    const float* p = ws + WS_POOL + (long)g * HIDDEN;
    float s = cls_b[k];
#pragma unroll 8
    for (int c = 0; c < HIDDEN; ++c) s += p[c] * cls_W[c * NUM_CLASSES + k];
    out[t] = s;
}

// ---------------------------------------------------------------------------
extern "C" void kernel_launch(void* const* d_in, const int* in_sizes, int n_in,
                              void* d_out, int out_size, void* d_ws, size_t ws_size,
                              hipStream_t stream) {
    const float* x        = (const float*)d_in[0];
    const int*   ei       = (const int*)  d_in[1];
    const int*   batch    = (const int*)  d_in[2];
    const float* W        = (const float*)d_in[3];
    const float* att_src  = (const float*)d_in[4];
    const float* att_dst  = (const float*)d_in[5];
    const float* bias     = (const float*)d_in[6];
    const float* cls_W    = (const float*)d_in[7];
    const float* cls_b    = (const float*)d_in[8];
    float* out = (float*)d_out;
    float* ws  = (float*)d_ws;
    (void)in_sizes; (void)n_in; (void)out_size; (void)ws_size;

    gat_init<<<2048, 256, 0, stream>>>(ws);

    dim3 ggrid(N_NODES / 16, HIDDEN / 16);     // 6250 x 12, exact
    gat_gemm_wmma<<<ggrid, 32, 0, stream>>>(x, W, ws);

    gat_logits<<<(N_NODES * HEADS + 255) / 256, 256, 0, stream>>>(att_src, att_dst, ws);

    gat_denom<<<(E_TOT + 255) / 256, 256, 0, stream>>>(ei, ws);

    long sc_threads = (long)E_TOT * 32;
    gat_scatter<<<(unsigned)((sc_threads + 255) / 256), 256, 0, stream>>>(ei, ws);

    long pl = (long)N_NODES * HIDDEN;
    gat_pool<<<(unsigned)((pl + 255) / 256), 256, 0, stream>>>(bias, batch, ws);

    gat_classify<<<1, 256, 0, stream>>>(cls_W, cls_b, ws, out);
}